// DotInteractionGather_62715112456554
// MI455X (gfx1250) — compile-verified
//
#include <hip/hip_runtime.h>

// DLRM dot-interaction + tril gather + concat, f32 via V_WMMA_F32_16X16X4_F32.
// B=65536, F=27, D=128, out row = 128 + 27*26/2 = 479 floats.

#define F_DIM 27
#define D_DIM 128
#define TRIL  351          // 27*26/2
#define OUT_COLS 479       // 128 + TRIL
#define BATCH_PER_BLOCK 8  // 8 waves of 32 per block

typedef __attribute__((ext_vector_type(2))) float v2f;
typedef __attribute__((ext_vector_type(8))) float v8f;

__global__ __launch_bounds__(256)
void dot_interaction_kernel(const float* __restrict__ features,
                            const float* __restrict__ bottom,
                            float* __restrict__ out,
                            int B)
{
    __shared__ float stage[BATCH_PER_BLOCK][TRIL + 1];

    const int lane = threadIdx.x & 31;
    const int wave = threadIdx.x >> 5;
    const int b = blockIdx.x * BATCH_PER_BLOCK + wave;
    if (b >= B) return;  // wave-uniform exit

    const float* fb = features + (size_t)b * (F_DIM * D_DIM);

    // f32 16x16x4 A/B operand layout (identical for A=MxK and B=KxN of A*A^T):
    //   lanes 0-15  hold (row = lane,    K = kbase+0, kbase+1) in v[0],v[1]
    //   lanes 16-31 hold (row = lane-16, K = kbase+2, kbase+3) in v[0],v[1]
    const int half = lane >> 4;
    const int r    = lane & 15;
    const int row1r = r + 16;                       // rows 16..31 (27..31 padded)
    const float pad = (row1r < F_DIM) ? 1.0f : 0.0f;
    const int row1  = (row1r < F_DIM) ? row1r : (F_DIM - 1);

    const float* p0 = fb + r    * D_DIM + 2 * half;
    const float* p1 = fb + row1 * D_DIM + 2 * half;

    v8f c00 = {};  // G[0:16, 0:16]
    v8f c10 = {};  // G[16:32, 0:16]
    v8f c11 = {};  // G[16:32,16:32]

    #pragma unroll 8
    for (int k = 0; k < D_DIM; k += 4) {
        v2f a0 = *(const v2f*)(p0 + k);
        v2f a1 = *(const v2f*)(p1 + k);
        a1.x *= pad;   // zero padded rows without touching EXEC
        a1.y *= pad;
        c00 = __builtin_amdgcn_wmma_f32_16x16x4_f32(false, a0, false, a0,
                                                    (short)0, c00, false, false);
        c10 = __builtin_amdgcn_wmma_f32_16x16x4_f32(false, a1, false, a0,
                                                    (short)0, c10, false, false);
        c11 = __builtin_amdgcn_wmma_f32_16x16x4_f32(false, a1, false, a1,
                                                    (short)0, c11, false, false);
    }

    // ---- scatter strict-lower-triangle into LDS staging ----
    // C/D layout: VGPR rr -> M = rr + 8*(lane>=16), N = lane&15
    float* st = stage[wave];
    const int mt = 8 * half;   // M offset within tile from lane half
    const int n0 = r;          // N within tile
    #pragma unroll
    for (int rr = 0; rr < 8; ++rr) {
        {   // tile (0,0)
            int m = rr + mt, n = n0;
            if (n < m) st[m * (m - 1) / 2 + n] = c00[rr];
        }
        {   // tile (1,0): always n < m
            int m = 16 + rr + mt, n = n0;
            if (m < F_DIM) st[m * (m - 1) / 2 + n] = c10[rr];
        }
        {   // tile (1,1)
            int m = 16 + rr + mt, n = 16 + n0;
            if (m < F_DIM && n < m) st[m * (m - 1) / 2 + n] = c11[rr];
        }
    }

    // cross-lane LDS dependency within the wave: drain DS ops
    asm volatile("s_wait_dscnt 0x0" ::: "memory");

    // ---- coalesced writeback ----
    float* ob = out + (size_t)b * OUT_COLS;
    const float* bb = bottom + (size_t)b * D_DIM;
    #pragma unroll
    for (int t = lane; t < D_DIM; t += 32)   // bottom_mlp_out passthrough
        ob[t] = bb[t];
    #pragma unroll
    for (int t = lane; t < TRIL; t += 32)    // tril(G) gather
        ob[D_DIM + t] = st[t];
}

extern "C" void kernel_launch(void* const* d_in, const int* in_sizes, int n_in,
                              void* d_out, int out_size, void* d_ws, size_t ws_size,
                              hipStream_t stream) {
    const float* features = (const float*)d_in[0];
    const float* bottom   = (const float*)d_in[1];
    float* out = (float*)d_out;

    const int B = in_sizes[0] / (F_DIM * D_DIM);   // 65536
    const int blocks = (B + BATCH_PER_BLOCK - 1) / BATCH_PER_BLOCK;
    dot_interaction_kernel<<<blocks, 256, 0, stream>>>(features, bottom, out, B);
}